// DistMultScore_1872605741811
// MI455X (gfx1250) — compile-verified
//
#include <hip/hip_runtime.h>

// DistMult edge scoring for MI455X (gfx1250, wave32).
//
// score[e] = sum_d node_emb[src[e],d] * rel_emb[e,d] * node_emb[dst[e],d], D=64.
//
// Bandwidth-bound (~300 MB HBM traffic, ~13 us floor at 23.3 TB/s).
// Strategy:
//   * 16 edges per wave, laid out directly in the V_WMMA_F32_16X16X4_F32
//     A-operand layout (lanes 0-15 = K0/K1, lanes 16-31 = K2/K3).
//   * Each lane privately accumulates its dim-partition with float4 loads
//     (lane<16: even quads, lane>=16: odd quads -> 32B contiguous per row
//     per iteration across the half-waves).
//   * ONE v_wmma_f32_16x16x4_f32 against an all-ones B reduces all 64 dims
//     for 16 edges at once: D[m][n] == score[m] in every column.
//   * Non-temporal loads on the 256 MB rel stream + NT stores on the output
//     so the 25.6 MB node table stays resident in the 192 MB L2.
//   * Tail edges: clamp the index (loads stay in-bounds); a clamped edge only
//     pollutes its own D-row, which is never stored. EXEC stays all-ones
//     through the WMMA as the ISA requires.

typedef __attribute__((ext_vector_type(2))) float v2f;
typedef __attribute__((ext_vector_type(4))) float v4f;
typedef __attribute__((ext_vector_type(8))) float v8f;

#define DIM               64
#define EDGES_PER_WAVE    16
#define WAVES_PER_BLOCK   8
#define THREADS_PER_BLOCK (WAVES_PER_BLOCK * 32)
#define EDGES_PER_BLOCK   (WAVES_PER_BLOCK * EDGES_PER_WAVE)

__global__ __launch_bounds__(THREADS_PER_BLOCK)
void distmult_wmma_kernel(const float* __restrict__ node_emb,
                          const float* __restrict__ rel_emb,
                          const int*   __restrict__ src,
                          const int*   __restrict__ dst,
                          float*       __restrict__ out,
                          int n_edges)
{
    const int lane = threadIdx.x & 31;
    const int wave = threadIdx.x >> 5;

    // Tile of 16 edges owned by this wave (uniform across the wave).
    const long long base =
        (long long)blockIdx.x * EDGES_PER_BLOCK + (long long)wave * EDGES_PER_WAVE;
    if (base >= n_edges) return;  // whole wave exits together; EXEC stays full

    const int m  = lane & 15;   // edge slot 0..15 (row M of the A matrix)
    const int hi = lane >> 4;   // 0: quads 0,2,..,14 (K0/K1)  1: quads 1,3,..,15 (K2/K3)

    long long e  = base + m;
    long long ec = (e < n_edges) ? e : (long long)(n_edges - 1);  // clamp tail

    const long long srow = (long long)src[ec] * DIM;
    const long long drow = (long long)dst[ec] * DIM;

    const float* __restrict__ hptr = node_emb + srow + 4 * hi;
    const float* __restrict__ tptr = node_emb + drow + 4 * hi;
    const float* __restrict__ rptr = rel_emb + ec * DIM + 4 * hi;

    // Private partial products over this lane's 32 dims (8 float4 quads).
    v4f acc4 = {0.f, 0.f, 0.f, 0.f};
#pragma unroll
    for (int q = 0; q < 8; ++q) {
        const int off = q * 8;  // elements: quad (2q + hi)
        v4f h = *(const v4f*)(hptr + off);                              // L2-resident gather
        v4f t = *(const v4f*)(tptr + off);                              // L2-resident gather
        v4f r = __builtin_nontemporal_load((const v4f*)(rptr + off));   // streamed once
        acc4 += h * r * t;
    }

    // Fold into this lane's two K-slots of the 16x4 f32 A operand.
    v2f a;
    a.x = acc4.x + acc4.y;  // K = 0 (lanes 0-15) / K = 2 (lanes 16-31)
    a.y = acc4.z + acc4.w;  // K = 1 (lanes 0-15) / K = 3 (lanes 16-31)

    v2f bones = {1.0f, 1.0f};          // all-ones 4x16 B
    v8f c = {};                        // zero C accumulator

    // D(16x16) = A(16x4) x ones(4x16): every column n holds score[m].
    c = __builtin_amdgcn_wmma_f32_16x16x4_f32(
            /*neg_a=*/false, a, /*neg_b=*/false, bones,
            /*c_mod=*/(short)0, c, /*reuse_a=*/false, /*reuse_b=*/false);

    // D layout: lane l, VGPR j -> D[j + 8*(l>=16)][l & 15].
    // Lanes 0-7 emit scores 0-7, lanes 16-23 emit scores 8-15.
    const int j  = lane & 7;
    float s01 = (j & 1) ? c[1] : c[0];
    float s23 = (j & 1) ? c[3] : c[2];
    float s45 = (j & 1) ? c[5] : c[4];
    float s67 = (j & 1) ? c[7] : c[6];
    float s03 = (j & 2) ? s23 : s01;
    float s47 = (j & 2) ? s67 : s45;
    float sc  = (j & 4) ? s47 : s03;

    const long long oidx = base + ((long long)hi << 3) + j;
    if (((lane & 15) < 8) && (oidx < n_edges)) {
        __builtin_nontemporal_store(sc, &out[oidx]);
    }
}

extern "C" void kernel_launch(void* const* d_in, const int* in_sizes, int n_in,
                              void* d_out, int out_size, void* d_ws, size_t ws_size,
                              hipStream_t stream) {
    const float* node_emb = (const float*)d_in[0];
    const float* rel_emb  = (const float*)d_in[1];
    const int*   src      = (const int*)d_in[2];   // harness contract: integer -> const int*
    const int*   dst      = (const int*)d_in[3];
    float*       out      = (float*)d_out;

    const int n_edges = in_sizes[2];  // element count of src == E

    const int grid = (n_edges + EDGES_PER_BLOCK - 1) / EDGES_PER_BLOCK;
    distmult_wmma_kernel<<<dim3(grid), dim3(THREADS_PER_BLOCK), 0, stream>>>(
        node_emb, rel_emb, src, dst, out, n_edges);
}